// GCN_66511863546567
// MI455X (gfx1250) — compile-verified
//
#include <hip/hip_runtime.h>
#include <hip/hip_bf16.h>

// GCN forward for MI455X (gfx1250). fp32 end-to-end; dense transforms use
// V_WMMA_F32_16X16X4_F32 (fp32 WMMA keeps reference precision; the GEMM is
// HBM-bound at ~205MB of x traffic so matrix-core rate is irrelevant).

typedef float v2f __attribute__((ext_vector_type(2)));
typedef float v8f __attribute__((ext_vector_type(8)));

#define NF   512
#define HID  8
#define NCLS 16

// ---------------- degree / norm ----------------
__global__ void k_fill1(float* __restrict__ deg, int n) {
    int i = blockIdx.x * blockDim.x + threadIdx.x;
    if (i < n) deg[i] = 1.0f;                       // self-loop contribution
}

__global__ void k_count(const int* __restrict__ dst, float* __restrict__ deg, int E) {
    int e = blockIdx.x * blockDim.x + threadIdx.x;
    if (e < E) atomicAdd(&deg[dst[e]], 1.0f);
}

__global__ void k_rsqrt(float* __restrict__ d, int n) {
    int i = blockIdx.x * blockDim.x + threadIdx.x;
    if (i < n) d[i] = rsqrtf(d[i]);                 // deg >= 1 always
}

// ---------------- GEMM1: h1 = x @ W1 (WMMA f32 16x16x4) ----------------
// One wave computes a 16-row tile over full K=512. W1 zero-padded to 512x16 in LDS.
__global__ void __launch_bounds__(256) k_gemm1(const float* __restrict__ x,
                                               const float* __restrict__ W1,
                                               float* __restrict__ h1,
                                               int n_nodes, int ntiles) {
    __shared__ float Wlds[NF * 16];
    const int tid = threadIdx.x;
    for (int idx = tid; idx < NF * 16; idx += 256) {
        int k = idx >> 4, n = idx & 15;
        Wlds[idx] = (n < HID) ? W1[k * HID + n] : 0.0f;
    }
    __syncthreads();

    const int wave = tid >> 5;
    const int lane = tid & 31;
    const int tile = blockIdx.x * 8 + wave;
    if (tile >= ntiles) return;                     // wave-uniform: EXEC stays all-ones

    const int m  = lane & 15;                       // A: row M ; B: col N
    const int hi = lane >> 4;                       // half-wave selects K+2/K+3
    const int base = tile * 16;
    int row = base + m;
    if (row >= n_nodes) row = n_nodes - 1;          // clamp loads, guard stores later
    const float* xrow = x + (size_t)row * NF;

    v8f c = {};
    for (int kk = 0; kk < NF; kk += 4) {
        const int k0 = kk + 2 * hi;
        v2f a; a.x = xrow[k0];            a.y = xrow[k0 + 1];        // one b64 load
        v2f b; b.x = Wlds[k0 * 16 + m];   b.y = Wlds[(k0 + 1) * 16 + m];
        c = __builtin_amdgcn_wmma_f32_16x16x4_f32(false, a, false, b,
                                                  (short)0, c, false, false);
        __builtin_prefetch(xrow + kk + 64, 0, 1);   // global_prefetch_b8 ahead
    }

    if (m < HID) {                                  // only 8 valid output columns
        #pragma unroll
        for (int j = 0; j < 8; ++j) {
            int r = base + j + 8 * hi;              // C/D layout: M = j + 8*hi
            if (r < n_nodes) h1[(size_t)r * HID + m] = c[j];
        }
    }
}

// ---------------- conv1 aggregation ----------------
__global__ void k_self1(const float* __restrict__ h1, const float* __restrict__ dinv,
                        float* __restrict__ agg1, int nelem) {
    int idx = blockIdx.x * blockDim.x + threadIdx.x;
    if (idx < nelem) {
        float di = dinv[idx >> 3];
        agg1[idx] = h1[idx] * di * di;              // self-loop message
    }
}

__global__ void k_scat1(const int* __restrict__ src, const int* __restrict__ dst,
                        const float* __restrict__ dinv, const float* __restrict__ h1,
                        float* __restrict__ agg1, int E) {
    int e = blockIdx.x * blockDim.x + threadIdx.x;
    if (e >= E) return;
    int s = src[e], d = dst[e];
    float w = dinv[s] * dinv[d];
    const float4* hp = (const float4*)(h1 + (size_t)s * HID);
    float4 h0 = hp[0], h4 = hp[1];
    float* o = agg1 + (size_t)d * HID;
    atomicAdd(o + 0, h0.x * w); atomicAdd(o + 1, h0.y * w);
    atomicAdd(o + 2, h0.z * w); atomicAdd(o + 3, h0.w * w);
    atomicAdd(o + 4, h4.x * w); atomicAdd(o + 5, h4.y * w);
    atomicAdd(o + 6, h4.z * w); atomicAdd(o + 7, h4.w * w);
}

// ---------------- GEMM2 fused: t2 = elu(agg1 + b1) @ W2 (WMMA) ----------------
__global__ void __launch_bounds__(256) k_gemm2(const float* __restrict__ agg1,
                                               const float* __restrict__ b1,
                                               const float* __restrict__ W2,
                                               float* __restrict__ t2,
                                               int n_nodes, int ntiles) {
    const int tid  = threadIdx.x;
    const int wave = tid >> 5;
    const int lane = tid & 31;
    const int tile = blockIdx.x * 8 + wave;
    if (tile >= ntiles) return;

    const int m  = lane & 15;
    const int hi = lane >> 4;
    const int base = tile * 16;
    int row = base + m;
    if (row >= n_nodes) row = n_nodes - 1;
    const float* arow = agg1 + (size_t)row * HID;

    v8f c = {};
    #pragma unroll
    for (int kk = 0; kk < HID; kk += 4) {
        const int k0 = kk + 2 * hi;
        float a0 = arow[k0]     + b1[k0];
        float a1 = arow[k0 + 1] + b1[k0 + 1];
        a0 = a0 > 0.0f ? a0 : (expf(a0) - 1.0f);    // ELU (elementwise ok in-fragment)
        a1 = a1 > 0.0f ? a1 : (expf(a1) - 1.0f);
        v2f a; a.x = a0; a.y = a1;
        v2f b; b.x = W2[k0 * NCLS + m]; b.y = W2[(k0 + 1) * NCLS + m];
        c = __builtin_amdgcn_wmma_f32_16x16x4_f32(false, a, false, b,
                                                  (short)0, c, false, false);
    }
    #pragma unroll
    for (int j = 0; j < 8; ++j) {
        int r = base + j + 8 * hi;
        if (r < n_nodes) t2[(size_t)r * NCLS + m] = c[j];
    }
}

// ---------------- conv2 aggregation (into d_out) ----------------
__global__ void k_self2(const float* __restrict__ t2, const float* __restrict__ dinv,
                        float* __restrict__ out, int nelem) {
    int idx = blockIdx.x * blockDim.x + threadIdx.x;
    if (idx < nelem) {
        float di = dinv[idx >> 4];
        out[idx] = t2[idx] * di * di;
    }
}

__global__ void k_scat2(const int* __restrict__ src, const int* __restrict__ dst,
                        const float* __restrict__ dinv, const float* __restrict__ t2,
                        float* __restrict__ out, int E) {
    int e = blockIdx.x * blockDim.x + threadIdx.x;
    if (e >= E) return;
    int s = src[e], d = dst[e];
    float w = dinv[s] * dinv[d];
    const float4* tp = (const float4*)(t2 + (size_t)s * NCLS);
    float* o = out + (size_t)d * NCLS;
    #pragma unroll
    for (int q = 0; q < 4; ++q) {
        float4 v = tp[q];
        atomicAdd(o + 4 * q + 0, v.x * w);
        atomicAdd(o + 4 * q + 1, v.y * w);
        atomicAdd(o + 4 * q + 2, v.z * w);
        atomicAdd(o + 4 * q + 3, v.w * w);
    }
}

// ---------------- bias + log_softmax (in place on d_out) ----------------
__global__ void k_lsm(float* __restrict__ out, const float* __restrict__ b2, int n) {
    int i = blockIdx.x * blockDim.x + threadIdx.x;
    if (i >= n) return;
    float* row = out + (size_t)i * NCLS;
    float v[NCLS];
    float mx = -INFINITY;
    #pragma unroll
    for (int j = 0; j < NCLS; ++j) { v[j] = row[j] + b2[j]; mx = fmaxf(mx, v[j]); }
    float s = 0.0f;
    #pragma unroll
    for (int j = 0; j < NCLS; ++j) s += expf(v[j] - mx);
    float l = logf(s);
    #pragma unroll
    for (int j = 0; j < NCLS; ++j) row[j] = v[j] - mx - l;
}

static inline size_t align128(size_t n) { return (n + 127) & ~(size_t)127; }

extern "C" void kernel_launch(void* const* d_in, const int* in_sizes, int n_in,
                              void* d_out, int out_size, void* d_ws, size_t ws_size,
                              hipStream_t stream) {
    const float* x  = (const float*)d_in[0];
    const int*   ei = (const int*)  d_in[1];   // (2, E) flat, int32 (JAX default no-x64)
    const float* W1 = (const float*)d_in[2];
    const float* b1 = (const float*)d_in[3];
    const float* W2 = (const float*)d_in[4];
    const float* b2 = (const float*)d_in[5];
    float* out = (float*)d_out;

    const int N = in_sizes[0] / NF;
    const int E = in_sizes[1] / 2;
    const int* src = ei;
    const int* dst = ei + E;

    float* ws = (float*)d_ws;
    size_t off = 0;
    float* dinv = ws + off; off += align128((size_t)N);
    float* h1   = ws + off; off += align128((size_t)N * HID);
    float* agg1 = ws + off; off += align128((size_t)N * HID);
    float* t2   = ws + off; off += align128((size_t)N * NCLS);
    (void)ws_size; (void)n_in; (void)out_size;

    const int BT = 256;
    const int ntiles = (N + 15) / 16;
    const int gtile  = (ntiles + 7) / 8;

    k_fill1 <<<(N + BT - 1) / BT, BT, 0, stream>>>(dinv, N);
    k_count <<<(E + BT - 1) / BT, BT, 0, stream>>>(dst, dinv, E);
    k_rsqrt <<<(N + BT - 1) / BT, BT, 0, stream>>>(dinv, N);

    k_gemm1 <<<gtile, BT, 0, stream>>>(x, W1, h1, N, ntiles);

    k_self1 <<<((N * HID) + BT - 1) / BT, BT, 0, stream>>>(h1, dinv, agg1, N * HID);
    k_scat1 <<<(E + BT - 1) / BT, BT, 0, stream>>>(src, dst, dinv, h1, agg1, E);

    k_gemm2 <<<gtile, BT, 0, stream>>>(agg1, b1, W2, t2, N, ntiles);

    k_self2 <<<((N * NCLS) + BT - 1) / BT, BT, 0, stream>>>(t2, dinv, out, N * NCLS);
    k_scat2 <<<(E + BT - 1) / BT, BT, 0, stream>>>(src, dst, dinv, t2, out, E);

    k_lsm   <<<(N + BT - 1) / BT, BT, 0, stream>>>(out, b2, N);
}